// CohesionGNN_2920577761970
// MI455X (gfx1250) — compile-verified
//
#include <hip/hip_runtime.h>
#include <hip/hip_bf16.h>
#include <math.h>

#define HID 32
#define NEG_SLOPE 0.2f
#define EPSF 1e-16f

typedef float v2f __attribute__((ext_vector_type(2)));
typedef float v8f __attribute__((ext_vector_type(8)));

// Float atomic max via signed/unsigned integer atomics (works for mixed signs,
// lowers to L2-side global_atomic_max_i32 / global_atomic_min_u32).
__device__ __forceinline__ void atomicMaxF(float* addr, float val) {
  if (val >= 0.0f) {
    atomicMax((int*)addr, __float_as_int(val));
  } else {
    atomicMin((unsigned int*)addr, __float_as_uint(val));
  }
}

__global__ void fill_kernel(float* __restrict__ p, int n, float v) {
  int i = blockIdx.x * blockDim.x + threadIdx.x;
  if (i < n) p[i] = v;
}

// w3 = W_edge @ att_edge  (3-vector), collapses the edge projection:
// (ea @ W_edge) @ att_edge == ea @ (W_edge @ att_edge).
__global__ void w3_kernel(const float* __restrict__ We, const float* __restrict__ ae,
                          float* __restrict__ w3) {
  int k = threadIdx.x;
  if (k < 3) {
    float s = 0.f;
    for (int j = 0; j < HID; ++j) s += We[k * HID + j] * ae[j];
    w3[k] = s;
  }
}

// Pass 1 over edges: in-degree + incoming edge-attr sums (for self-loop 'mean' fill).
__global__ void edge_deg_kernel(const int* __restrict__ ei, const float* __restrict__ ea,
                                float* __restrict__ deg, float* __restrict__ lsum, int E) {
  int e = blockIdx.x * blockDim.x + threadIdx.x;
  if (e >= E) return;
  int dst = ei[E + e];
  atomicAdd(&deg[dst], 1.0f);
  atomicAdd(&lsum[dst * 3 + 0], ea[e * 3 + 0]);
  atomicAdd(&lsum[dst * 3 + 1], ea[e * 3 + 1]);
  atomicAdd(&lsum[dst * 3 + 2], ea[e * 3 + 2]);
}

// h = x @ W : [N,12] @ [12,32] -> [N,32] using V_WMMA_F32_16X16X4_F32.
// One wave per 16-row tile; K=12 is exactly 3 steps of K=4; two N-halves (cols 0-15, 16-31).
// A 16x4 f32 layout: lanes 0-15 hold (K=k0,k0+1), lanes 16-31 hold (K=k0+2,k0+3), M = lane&15.
// B 4x16 f32 layout (mirror): lane half selects K pair, N = lane&15.
// C/D 16x16 f32: lane<16 -> N=lane, rows 0..7 in c[0..7]; lane>=16 -> N=lane-16, rows 8..15.
//
// Tail handling: A rows are clamped to N-1 at load time, so out-of-range output
// rows hold values identical to row N-1's outputs. Clamping the STORE address is
// therefore value-identical (benign duplicate store of the same value) -> no
// per-row exec-mask branches needed; the epilogue is one straight store clause.
__global__ void h_wmma_kernel(const float* __restrict__ x, const float* __restrict__ W,
                              float* __restrict__ h, int N) {
#if defined(__gfx1250__) && __has_builtin(__builtin_amdgcn_wmma_f32_16x16x4_f32)
  int lane = threadIdx.x;
  int half = lane >> 4;   // 0: K pair {0,1}; 1: K pair {2,3}
  int l    = lane & 15;
  int r0   = blockIdx.x * 16;

  v8f c0 = {0.f, 0.f, 0.f, 0.f, 0.f, 0.f, 0.f, 0.f};
  v8f c1 = {0.f, 0.f, 0.f, 0.f, 0.f, 0.f, 0.f, 0.f};

  int row = r0 + l;
  if (row >= N) row = N - 1;  // clamp loads; store clamp below is value-identical

#pragma unroll
  for (int k0 = 0; k0 < 12; k0 += 4) {
    int kk = k0 + half * 2;
    v2f a, b0, b1;
    a.x  = x[row * 12 + kk];
    a.y  = x[row * 12 + kk + 1];
    b0.x = W[kk * HID + l];
    b0.y = W[(kk + 1) * HID + l];
    b1.x = W[kk * HID + 16 + l];
    b1.y = W[(kk + 1) * HID + 16 + l];
    c0 = __builtin_amdgcn_wmma_f32_16x16x4_f32(false, a, false, b0, (short)0, c0, false, false);
    c1 = __builtin_amdgcn_wmma_f32_16x16x4_f32(false, a, false, b1, (short)0, c1, false, false);
  }

  int rbase = r0 + half * 8;
#pragma unroll
  for (int m = 0; m < 8; ++m) {
    int r = rbase + m;
    if (r >= N) r = N - 1;  // clamped duplicate store of identical value
    h[r * HID + l]      = c0[m];
    h[r * HID + l + 16] = c1[m];
  }
#else
  // Scalar fallback (also what the host-side pass parses).
  int l  = threadIdx.x;
  int r0 = blockIdx.x * 16;
  for (int mr = 0; mr < 16; ++mr) {
    int row = r0 + mr;
    if (row >= N) break;
    float s = 0.f;
    for (int k = 0; k < 12; ++k) s += x[row * 12 + k] * W[k * HID + l];
    h[row * HID + l] = s;
  }
#endif
}

// Per-node: attention logit halves, self-loop alpha, init m[i] = alpha_loop.
__global__ void node_prep_kernel(const float* __restrict__ h, const float* __restrict__ att_src,
                                 const float* __restrict__ att_dst, const float* __restrict__ w3,
                                 const float* __restrict__ deg, const float* __restrict__ lsum,
                                 float* __restrict__ a_src, float* __restrict__ a_dst,
                                 float* __restrict__ alpha_l, float* __restrict__ m, int N) {
  int i = blockIdx.x * blockDim.x + threadIdx.x;
  if (i >= N) return;
  float s = 0.f, d = 0.f;
#pragma unroll
  for (int j = 0; j < HID; ++j) {
    float hv = h[i * HID + j];
    s += hv * att_src[j];
    d += hv * att_dst[j];
  }
  a_src[i] = s;
  a_dst[i] = d;
  float dg = fmaxf(deg[i], 1.0f);
  float ae = (lsum[i * 3 + 0] * w3[0] + lsum[i * 3 + 1] * w3[1] + lsum[i * 3 + 2] * w3[2]) / dg;
  float al = s + d + ae;
  al = (al > 0.f) ? al : NEG_SLOPE * al;
  alpha_l[i] = al;
  m[i] = al;  // every node has a self-loop -> valid segment-max init
}

// Pass 2 over edges: leaky-relu logit, segment max.
__global__ void edge_alpha_kernel(const int* __restrict__ ei, const float* __restrict__ ea,
                                  const float* __restrict__ w3, const float* __restrict__ a_src,
                                  const float* __restrict__ a_dst, float* __restrict__ alphaE,
                                  float* __restrict__ m, int E) {
  int e = blockIdx.x * blockDim.x + threadIdx.x;
  if (e >= E) return;
  int src = ei[e];
  int dst = ei[E + e];
  float ae = ea[e * 3 + 0] * w3[0] + ea[e * 3 + 1] * w3[1] + ea[e * 3 + 2] * w3[2];
  float al = a_src[src] + a_dst[dst] + ae;
  al = (al > 0.f) ? al : NEG_SLOPE * al;
  alphaE[e] = al;
  atomicMaxF(&m[dst], al);
}

// Pass 3 over edges (4 threads/edge, 8 cols each): e = exp(alpha - m), scatter z and h*e.
// Softmax denominator is folded: sum(h*e)/(z+eps) == sum(h * e/(z+eps)).
__global__ void edge_accum_kernel(const int* __restrict__ ei, const float* __restrict__ h,
                                  const float* __restrict__ alphaE, const float* __restrict__ m,
                                  float* __restrict__ z, float* __restrict__ out_acc, int E) {
  int t = blockIdx.x * blockDim.x + threadIdx.x;
  int e = t >> 2;
  if (e >= E) return;
  int part = t & 3;
  int src = ei[e];
  int dst = ei[E + e];
  float ex = expf(alphaE[e] - m[dst]);
  if (part == 0) atomicAdd(&z[dst], ex);
  const float* hs = &h[src * HID + part * 8];
  float* oa = &out_acc[dst * HID + part * 8];
#pragma unroll
  for (int j = 0; j < 8; ++j) atomicAdd(&oa[j], hs[j] * ex);
}

// Per-node: merge self-loop, normalize, bias+ReLU (out stored in-place over out_acc),
// gate logit, segment max over graphs.
__global__ void node_finalize_kernel(const float* __restrict__ h, const float* __restrict__ alpha_l,
                                     const float* __restrict__ m, const float* __restrict__ z,
                                     const float* __restrict__ bias, const float* __restrict__ W_gate,
                                     const float* __restrict__ b_gate, const int* __restrict__ batch,
                                     float* __restrict__ out_acc, float* __restrict__ gate_raw,
                                     float* __restrict__ gm, int N) {
  int i = blockIdx.x * blockDim.x + threadIdx.x;
  if (i >= N) return;
  float el = expf(alpha_l[i] - m[i]);
  float zz = z[i] + el + EPSF;
  float g = 0.f;
#pragma unroll
  for (int j = 0; j < HID; ++j) {
    float o = (out_acc[i * HID + j] + h[i * HID + j] * el) / zz + bias[j];
    o = fmaxf(o, 0.f);
    out_acc[i * HID + j] = o;  // now holds relu'd GAT output
    g += o * W_gate[j];
  }
  g += b_gate[0];
  gate_raw[i] = g;
  atomicMaxF(&gm[batch[i]], g);
}

// Per-node: gate exp, pooled accumulation.
__global__ void node_pool_kernel(const int* __restrict__ batch, const float* __restrict__ gate_raw,
                                 const float* __restrict__ gm, const float* __restrict__ out,
                                 float* __restrict__ gz, float* __restrict__ pooled, int N) {
  int i = blockIdx.x * blockDim.x + threadIdx.x;
  if (i >= N) return;
  int b = batch[i];
  float ge = expf(gate_raw[i] - gm[b]);
  atomicAdd(&gz[b], ge);
  float* pb = &pooled[b * HID];
  const float* oi = &out[i * HID];
#pragma unroll
  for (int j = 0; j < HID; ++j) atomicAdd(&pb[j], oi[j] * ge);
}

// Per-graph: normalize pooled, final linear + sigmoid. Empty graphs: gz=0, pooled=0
// -> sigmoid(b_out) = 0.5, matching JAX segment semantics.
__global__ void graph_out_kernel(const float* __restrict__ pooled, const float* __restrict__ gz,
                                 const float* __restrict__ W_out, const float* __restrict__ b_out,
                                 float* __restrict__ out, int G) {
  int g = blockIdx.x * blockDim.x + threadIdx.x;
  if (g >= G) return;
  float inv = 1.0f / (gz[g] + EPSF);
  float s = 0.f;
#pragma unroll
  for (int j = 0; j < HID; ++j) s += pooled[g * HID + j] * inv * W_out[j];
  s += b_out[0];
  out[g] = 1.0f / (1.0f + expf(-s));
}

extern "C" void kernel_launch(void* const* d_in, const int* in_sizes, int n_in,
                              void* d_out, int out_size, void* d_ws, size_t ws_size,
                              hipStream_t stream) {
  const float* x        = (const float*)d_in[0];
  const int*   ei       = (const int*)d_in[1];
  const float* ea       = (const float*)d_in[2];
  const int*   batch    = (const int*)d_in[3];
  const float* W        = (const float*)d_in[4];
  const float* att_src  = (const float*)d_in[5];
  const float* att_dst  = (const float*)d_in[6];
  const float* W_edge   = (const float*)d_in[7];
  const float* att_edge = (const float*)d_in[8];
  const float* bias     = (const float*)d_in[9];
  const float* W_gate   = (const float*)d_in[10];
  const float* b_gate   = (const float*)d_in[11];
  const float* W_out    = (const float*)d_in[12];
  const float* b_out    = (const float*)d_in[13];
  float* out = (float*)d_out;

  const int N = in_sizes[0] / 12;
  const int E = in_sizes[1] / 2;
  const int G = out_size;

  float* ws = (float*)d_ws;
  size_t off = 0;
  float* h        = ws + off; off += (size_t)N * HID;
  float* out_acc  = ws + off; off += (size_t)N * HID;
  float* alphaE   = ws + off; off += (size_t)E;
  float* a_src    = ws + off; off += N;
  float* a_dst    = ws + off; off += N;
  float* degv     = ws + off; off += N;
  float* lsum     = ws + off; off += (size_t)N * 3;
  float* alpha_l  = ws + off; off += N;
  float* mv       = ws + off; off += N;
  float* zv       = ws + off; off += N;
  float* gate_raw = ws + off; off += N;
  float* gm       = ws + off; off += G;
  float* gz       = ws + off; off += G;
  float* pooled   = ws + off; off += (size_t)G * HID;
  float* w3       = ws + off; off += 4;

  auto nb = [](long n, int b) { return (int)((n + b - 1) / b); };

  fill_kernel<<<nb((long)N * HID, 256), 256, 0, stream>>>(out_acc, N * HID, 0.f);
  fill_kernel<<<nb(N, 256), 256, 0, stream>>>(degv, N, 0.f);
  fill_kernel<<<nb((long)N * 3, 256), 256, 0, stream>>>(lsum, N * 3, 0.f);
  fill_kernel<<<nb(N, 256), 256, 0, stream>>>(zv, N, 0.f);
  fill_kernel<<<nb(G, 256), 256, 0, stream>>>(gz, G, 0.f);
  fill_kernel<<<nb((long)G * HID, 256), 256, 0, stream>>>(pooled, G * HID, 0.f);
  fill_kernel<<<nb(G, 256), 256, 0, stream>>>(gm, G, -INFINITY);

  w3_kernel<<<1, 4, 0, stream>>>(W_edge, att_edge, w3);
  edge_deg_kernel<<<nb(E, 256), 256, 0, stream>>>(ei, ea, degv, lsum, E);
  h_wmma_kernel<<<nb(N, 16), 32, 0, stream>>>(x, W, h, N);
  node_prep_kernel<<<nb(N, 256), 256, 0, stream>>>(h, att_src, att_dst, w3, degv, lsum,
                                                   a_src, a_dst, alpha_l, mv, N);
  edge_alpha_kernel<<<nb(E, 256), 256, 0, stream>>>(ei, ea, w3, a_src, a_dst, alphaE, mv, E);
  edge_accum_kernel<<<nb((long)E * 4, 256), 256, 0, stream>>>(ei, h, alphaE, mv, zv, out_acc, E);
  node_finalize_kernel<<<nb(N, 256), 256, 0, stream>>>(h, alpha_l, mv, zv, bias, W_gate, b_gate,
                                                       batch, out_acc, gate_raw, gm, N);
  node_pool_kernel<<<nb(N, 256), 256, 0, stream>>>(batch, gate_raw, gm, out_acc, gz, pooled, N);
  graph_out_kernel<<<nb(G, 64), 64, 0, stream>>>(pooled, gz, W_out, b_out, out, G);
}